// TriangularSylvesterNeRF_33586644255083
// MI455X (gfx1250) — compile-verified
//
#include <hip/hip_runtime.h>
#include <hip/hip_bf16.h>

typedef float v2f __attribute__((ext_vector_type(2)));
typedef float v8f __attribute__((ext_vector_type(8)));

#define ZDIM 32
#define HDIM 128
#define TBATCH 64                    // samples per workgroup
#define NWAVES 16
#define NTHREADS 512
#define H_S 80                       // padded sample stride for h staging [c][s]
#define FD_BASE (HDIM * H_S)         // 10240 floats
#define FD_S 1058                    // per-sample stride of full_d tile (32*33 padded + bank skew)
#define FD_ROW 33                    // padded row stride (j dimension)
#define SM_STRIDE (TBATCH * HDIM)    // 8192 floats per small-output matrix
#define LDS_FLOATS (FD_BASE + TBATCH * FD_S)   // 77952 floats = 311,808 B  (< 320 KB)

__device__ __forceinline__ float rlane(float x, int i) {
  return __int_as_float(__builtin_amdgcn_readlane(__float_as_int(x), i));
}

__device__ __forceinline__ v8f wmma_f32(v2f a, v2f b, v8f c) {
  // V_WMMA_F32_16X16X4_F32 : D = A(16x4) * B(4x16) + C(16x16)
  return __builtin_amdgcn_wmma_f32_16x16x4_f32(false, a, false, b, (short)0, c, false, false);
}

__global__ __launch_bounds__(NTHREADS)
void sylvester_flow_kernel(const float* __restrict__ z0,
                           const float* __restrict__ h,
                           const float* __restrict__ Wd,  const float* __restrict__ bd,
                           const float* __restrict__ Wd1, const float* __restrict__ bd1,
                           const float* __restrict__ Wd2, const float* __restrict__ bd2,
                           const float* __restrict__ Wb,  const float* __restrict__ bb,
                           float* __restrict__ out_z, float* __restrict__ out_ld)
{
  __shared__ float sh[LDS_FLOATS];

  const int tid   = threadIdx.x;
  const int lane  = tid & 31;
  const int wave  = tid >> 5;
  const int mrow  = lane & 15;   // WMMA row / column index within half-wave
  const int mhalf = lane >> 4;   // 0: K=0,1 / rows 0-7 ; 1: K=2,3 / rows 8-15
  const int mt     = wave & 3;   // sample row-tile (16 samples each)
  const int ngroup = wave >> 2;  // column-tile group
  const int s0 = blockIdx.x * TBATCH;

  // ---- 1) stage h[TB][H] into LDS transposed: sh[c*H_S + s] --------------
  #pragma unroll
  for (int r = 0; r < (TBATCH * HDIM) / NTHREADS; ++r) {
    int idx = r * NTHREADS + tid;
    int s = idx >> 7;
    int c = idx & 127;
    sh[c * H_S + s] = h[(size_t)(s0 + s) * HDIM + c];
  }
  __syncthreads();

  // ---- A-matrix cache: 16x4 fp32 tiles for this wave's row block, all 32 h-steps
  const int s_loc = mt * 16 + mrow;
  v2f acache[32];
  #pragma unroll
  for (int hh = 0; hh < 32; ++hh) {
    int kk = hh * 4 + 2 * mhalf;
    v2f a;
    a.x = sh[kk * H_S + s_loc];
    a.y = sh[(kk + 1) * H_S + s_loc];
    acache[hh] = a;
  }

  // ---- 2) small GEMMs: diag1, diag2, b -> sh[FD_BASE + mat*SM_STRIDE + s*128 + c']
  #pragma unroll 1
  for (int q = 0; q < 6; ++q) {
    int id2 = ngroup * 6 + q;            // 0..23 = (mat, ntile)
    int mat = id2 >> 3;                  // 0:Wd1  1:Wd2  2:Wb
    int nt2 = id2 & 7;
    const float* Wm = (mat == 0) ? Wd1 : (mat == 1) ? Wd2 : Wb;
    const float* bm = (mat == 0) ? bd1 : (mat == 1) ? bd2 : bb;
    int cc = nt2 * 16 + mrow;
    float bias = bm[cc];
    v8f acc;
    #pragma unroll
    for (int r = 0; r < 8; ++r) acc[r] = bias;
    const float* wrow = Wm + (size_t)cc * HDIM;
    v2f bmat[32];
    #pragma unroll
    for (int hh = 0; hh < 32; ++hh)
      bmat[hh] = *reinterpret_cast<const v2f*>(wrow + hh * 4 + 2 * mhalf);
    #pragma unroll
    for (int hh = 0; hh < 32; ++hh)
      acc = wmma_f32(acache[hh], bmat[hh], acc);
    int sbase = mt * 16 + mhalf * 8;
    #pragma unroll
    for (int r = 0; r < 8; ++r) {
      float v = acc[r];
      if (mat < 2) v = tanhf(v);
      sh[FD_BASE + mat * SM_STRIDE + (sbase + r) * HDIM + cc] = v;
    }
  }
  __syncthreads();

  // ---- 3) pull flow state into registers: this wave owns 4 samples -------
  const int sflow = wave * 4;
  float zr[4], ldr[4], d1r[16], d2r[16], br[16];
  #pragma unroll
  for (int si = 0; si < 4; ++si) {
    zr[si]  = z0[(size_t)(s0 + sflow + si) * ZDIM + lane];
    ldr[si] = 0.0f;
    #pragma unroll
    for (int k = 0; k < 4; ++k) {
      int coff = lane * 4 + k;     // channel c' = j*4 + k
      d1r[si * 4 + k] = sh[FD_BASE + 0 * SM_STRIDE + (sflow + si) * HDIM + coff];
      d2r[si * 4 + k] = sh[FD_BASE + 1 * SM_STRIDE + (sflow + si) * HDIM + coff];
      br [si * 4 + k] = sh[FD_BASE + 2 * SM_STRIDE + (sflow + si) * HDIM + coff];
    }
  }

  // ---- 4) fused per-k: GEMM full_d[:,:,:,k] slice -> LDS, then flow step -
  #pragma unroll 1
  for (int k = 0; k < 4; ++k) {
    __syncthreads();   // previous consumers of the fd/small region are done

    // GEMM: this wave computes 16 of the 64 column tiles (M=64, N=1024, K=128)
    #pragma unroll 1
    for (int tt = 0; tt < 16; ++tt) {
      int nt = ngroup * 16 + tt;
      int i_ = nt >> 1;                 // Z row index i
      int j0 = (nt & 1) << 4;           // Z col base
      int ccol = i_ * 128 + (j0 + mrow) * 4 + k;   // Wd row = channel index
      float bias = bd[ccol];
      v8f acc;
      #pragma unroll
      for (int r = 0; r < 8; ++r) acc[r] = bias;
      const float* wrow = Wd + (size_t)ccol * HDIM;
      v2f bmat[32];
      #pragma unroll
      for (int hh = 0; hh < 32; ++hh)
        bmat[hh] = *reinterpret_cast<const v2f*>(wrow + hh * 4 + 2 * mhalf);
      #pragma unroll
      for (int hh = 0; hh < 32; ++hh)
        acc = wmma_f32(acache[hh], bmat[hh], acc);
      int sb = mt * 16 + mhalf * 8;
      int base = FD_BASE + sb * FD_S + i_ * FD_ROW + j0 + mrow;
      #pragma unroll
      for (int r = 0; r < 8; ++r)
        sh[base + r * FD_S] = acc[r];
    }
    __syncthreads();

    // Flow step k for this wave's 4 samples (lane == Z index)
    const bool flip = (k & 1);
    #pragma unroll 1
    for (int si = 0; si < 4; ++si) {
      float zv = zr[si];
      float zper = flip ? __shfl_xor(zv, 31, 32) : zv;   // z[::-1] on odd k
      int fdb = FD_BASE + (sflow + si) * FD_S;
      // pre[j] = sum_{i>j} z_per[i]*fd[i,j] + z_per[j]*diag2[j] + b[j]
      float pre = br[si * 4 + k] + zper * d2r[si * 4 + k];
      #pragma unroll
      for (int i = 1; i < 32; ++i) {
        float zi = rlane(zper, i);
        float v  = sh[fdb + i * FD_ROW + lane];
        pre += (i > lane) ? zi * v : 0.0f;
      }
      float t = tanhf(pre);
      // dz[i] = sum_{j>i} t[j]*fd[i,j] + t[i]*diag1[i]
      float dz = t * d1r[si * 4 + k];
      #pragma unroll
      for (int j = 1; j < 32; ++j) {
        float tj = rlane(t, j);
        float v  = sh[fdb + lane * FD_ROW + j];
        dz += (j > lane) ? tj * v : 0.0f;
      }
      if (flip) dz = __shfl_xor(dz, 31, 32);
      zr[si] = zv + dz;
      float dj = (1.0f - t * t) * (d1r[si * 4 + k] * d2r[si * 4 + k]) + 1.0f;
      ldr[si] += logf(fabsf(dj));
    }
  }

  // ---- 5) outputs --------------------------------------------------------
  #pragma unroll
  for (int si = 0; si < 4; ++si) {
    int s = s0 + sflow + si;
    out_z[(size_t)s * ZDIM + lane] = zr[si];
    float v = ldr[si];
    #pragma unroll
    for (int off = 16; off > 0; off >>= 1) v += __shfl_xor(v, off, 32);
    if (lane == 0) out_ld[s] = v;
  }
}

extern "C" void kernel_launch(void* const* d_in, const int* in_sizes, int n_in,
                              void* d_out, int out_size, void* d_ws, size_t ws_size,
                              hipStream_t stream) {
  const float* z0  = (const float*)d_in[0];
  const float* h   = (const float*)d_in[1];
  const float* Wd  = (const float*)d_in[2];
  const float* bd  = (const float*)d_in[3];
  const float* Wd1 = (const float*)d_in[4];
  const float* bd1 = (const float*)d_in[5];
  const float* Wd2 = (const float*)d_in[6];
  const float* bd2 = (const float*)d_in[7];
  const float* Wb  = (const float*)d_in[8];
  const float* bb  = (const float*)d_in[9];
  // d_in[10] = is_test (unused)

  const int B = in_sizes[0] / ZDIM;      // 32768
  float* out   = (float*)d_out;
  float* out_z = out;                    // (B, 32)
  float* out_ld = out + (size_t)B * ZDIM;  // (B,)

  dim3 grid(B / TBATCH), block(NTHREADS);
  hipLaunchKernelGGL(sylvester_flow_kernel, grid, block, 0, stream,
                     z0, h, Wd, bd, Wd1, bd1, Wd2, bd2, Wb, bb, out_z, out_ld);
}